// BeliveMapsNMS_37005438222881
// MI455X (gfx1250) — compile-verified
//
#include <hip/hip_runtime.h>
#include <hip/hip_bf16.h>
#include <stdint.h>

// ---------------------------------------------------------------------------
// BeliveMapsNMS: 7x7 same-padded maxpool peak detection on [32,4,512,512] f32.
// Memory-bound (~640 MiB traffic). CDNA5 paths used:
//   - GLOBAL_LOAD_ASYNC_TO_LDS_B128 (ASYNCcnt + s_wait_asynccnt) for haloed
//     tile staging straight into LDS (no VGPR round-trip)
//   - non-temporal output stores so the 384 MiB write stream doesn't evict
//     the L2-resident input between the two passes
// ---------------------------------------------------------------------------

#define BM_W   512
#define BM_H   512
#define BM_P   128            // 32*4 planes
#define ROWS   8              // output rows per block
#define TROWS  (ROWS + 6)     // input rows incl. +/-3 halo
#define WPAD   (BM_W + 8)     // 4 pad cols each side -> 16B-aligned data start

typedef int   v4i __attribute__((ext_vector_type(4)));
typedef float v4f __attribute__((ext_vector_type(4)));     // native vector (NT-store OK)
typedef __attribute__((address_space(1))) v4i glb_v4i_t;   // global int4*
typedef __attribute__((address_space(3))) v4i lds_v4i_t;   // LDS int4*

#if defined(__has_builtin)
#if __has_builtin(__builtin_amdgcn_global_load_async_to_lds_b128)
#define HAVE_ASYNC_LDS 1
#endif
#endif
#ifndef HAVE_ASYNC_LDS
#define HAVE_ASYNC_LDS 0
#warning "gfx1250 async global->LDS builtin NOT available; using VGPR-staged fallback"
#endif

__device__ __forceinline__ void async_copy16(const float* g, float* l) {
#if HAVE_ASYNC_LDS
  // per-lane 16B global -> LDS copy, tracked by ASYNCcnt
  __builtin_amdgcn_global_load_async_to_lds_b128(
      (glb_v4i_t*)(uintptr_t)g,
      (lds_v4i_t*)(unsigned int)(uintptr_t)l,    // low 32 bits == LDS byte addr
      0, 0);
#else
  *(v4f*)l = *(const v4f*)g;
#endif
}

__device__ __forceinline__ void wait_async_lds() {
#if HAVE_ASYNC_LDS
#if defined(__has_builtin) && __has_builtin(__builtin_amdgcn_s_wait_asynccnt)
  __builtin_amdgcn_s_wait_asynccnt(0);
#else
  asm volatile("s_wait_asynccnt 0" ::: "memory");
#endif
#endif
}

// ---------------------------------------------------------------------------
// Kernel 1: per-(b,c)-plane global max -> d_ws[plane]
// ---------------------------------------------------------------------------
__global__ __launch_bounds__(1024) void bm_plane_max(const float* __restrict__ in,
                                                     float* __restrict__ vmax) {
  const int plane = blockIdx.x;
  const v4f* p = (const v4f*)(in + (size_t)plane * (BM_H * BM_W));
  float m = -INFINITY;
  for (int i = threadIdx.x; i < (BM_H * BM_W) / 4; i += 1024) {
    v4f v = p[i];
    m = fmaxf(m, fmaxf(fmaxf(v[0], v[1]), fmaxf(v[2], v[3])));
  }
  __shared__ float red[1024];
  red[threadIdx.x] = m;
  __syncthreads();
  for (int s = 512; s > 0; s >>= 1) {
    if (threadIdx.x < s) red[threadIdx.x] = fmaxf(red[threadIdx.x], red[threadIdx.x + s]);
    __syncthreads();
  }
  if (threadIdx.x == 0) vmax[plane] = red[0];
}

// ---------------------------------------------------------------------------
// Kernel 2: tiled separable 7x7 max + thresholds + 3 dense outputs
// grid = 128 planes * 64 row-blocks, 256 threads
// ---------------------------------------------------------------------------
__global__ __launch_bounds__(256) void bm_nms(const float* __restrict__ in,
                                              const float* __restrict__ vmaxbuf,
                                              float* __restrict__ out) {
  __shared__ float raw[TROWS * WPAD];   // haloed input tile
  __shared__ float rmx[TROWS * WPAD];   // horizontal 7-tap max

  const int tid    = threadIdx.x;
  const int plane  = blockIdx.x >> 6;
  const int rowblk = blockIdx.x & 63;
  const int r0     = rowblk * ROWS - 3;               // first (possibly <0) input row
  const size_t pbase = (size_t)plane * (BM_H * BM_W);
  const float* gplane = in + pbase;

  // left/right -inf pads: cols 0..3 and 516..519 of each tile row
  for (int i = tid; i < TROWS * 8; i += 256) {
    int r = i >> 3, c = i & 7;
    int col = (c < 4) ? c : (BM_W + c);               // 0..3 or 516..519
    raw[r * WPAD + col] = -INFINITY;
  }
  // body: 14 rows x 128 float4 = 1792 async 16B copies
  for (int i = tid; i < TROWS * (BM_W / 4); i += 256) {
    int r  = i >> 7;                                  // /128
    int c4 = i & 127;
    int gr = r0 + r;
    float* ldst = &raw[r * WPAD + 4 + c4 * 4];
    if (gr >= 0 && gr < BM_H) {
      async_copy16(gplane + (size_t)gr * BM_W + c4 * 4, ldst);
    } else {
      ldst[0] = ldst[1] = ldst[2] = ldst[3] = -INFINITY;
    }
  }
  wait_async_lds();
  __syncthreads();

  // horizontal 7-tap max: center global col x lives at LDS col x+4
  for (int i = tid; i < TROWS * BM_W; i += 256) {
    int r = i >> 9, x = i & 511;
    const float* row = &raw[r * WPAD + x + 1];        // cols x+1 .. x+7
    float m = row[0];
    m = fmaxf(m, row[1]); m = fmaxf(m, row[2]); m = fmaxf(m, row[3]);
    m = fmaxf(m, row[4]); m = fmaxf(m, row[5]); m = fmaxf(m, row[6]);
    rmx[r * WPAD + 4 + x] = m;
  }
  __syncthreads();

  const float vmax   = vmaxbuf[plane];
  const float relthr = 0.05f * vmax;
  const size_t Nall  = (size_t)BM_P * BM_H * BM_W;    // 33,554,432
  float* out_mask = out;
  float* out_abs  = out + Nall;
  float* out_rel  = out + 2 * Nall;

  // vertical 7-tap max + thresholds, non-temporal 16B stores
  for (int i = tid; i < ROWS * (BM_W / 4); i += 256) {
    int rr = i >> 7, c4 = i & 127;
    int grow = rowblk * ROWS + rr;
    v4f vmsk, vabs, vrel;
#pragma unroll
    for (int j = 0; j < 4; ++j) {
      int x = c4 * 4 + j;
      float xv = raw[(rr + 3) * WPAD + 4 + x];        // center value
      float mx = rmx[rr * WPAD + 4 + x];
#pragma unroll
      for (int k = 1; k < 7; ++k) mx = fmaxf(mx, rmx[(rr + k) * WPAD + 4 + x]);
      bool msk = (mx == xv) && (xv > 0.2f) && (xv > relthr);
      float sa = msk ? xv : 0.0f;
      vmsk[j] = msk ? 1.0f : 0.0f;
      vabs[j] = sa;
      vrel[j] = sa / vmax;
    }
    size_t idx = pbase + (size_t)grow * BM_W + c4 * 4;
    __builtin_nontemporal_store(vmsk, (v4f*)(out_mask + idx));
    __builtin_nontemporal_store(vabs, (v4f*)(out_abs + idx));
    __builtin_nontemporal_store(vrel, (v4f*)(out_rel + idx));
  }
}

extern "C" void kernel_launch(void* const* d_in, const int* in_sizes, int n_in,
                              void* d_out, int out_size, void* d_ws, size_t ws_size,
                              hipStream_t stream) {
  const float* in = (const float*)d_in[0];
  float* vmax = (float*)d_ws;                         // 128 floats of scratch
  float* out  = (float*)d_out;

  bm_plane_max<<<BM_P, 1024, 0, stream>>>(in, vmax);
  bm_nms<<<BM_P * (BM_H / ROWS), 256, 0, stream>>>(in, vmax, out);
}